// EGNN_73512660238857
// MI455X (gfx1250) — compile-verified
//
#include <hip/hip_runtime.h>

typedef float v2f __attribute__((ext_vector_type(2)));
typedef float v8f __attribute__((ext_vector_type(8)));

#define NNODE 4096
#define CH 64
#define NB 4

__device__ __forceinline__ v8f wmma4(v2f a, v2f b, v8f c) {
  // D = A(16x4) * B(4x16) + C, f32, wave32
  return __builtin_amdgcn_wmma_f32_16x16x4_f32(false, a, false, b, (short)0, c,
                                               false, false);
}

// ---------------------------------------------------------------------------
// Kernel 1: out = X @ W1, stored TRANSPOSED as outT[b][c][j] so the main
// kernel's B-fragments are contiguous float2 loads along j.
// One wave per (b, 16-row j tile).
// ---------------------------------------------------------------------------
__global__ __launch_bounds__(128) void k1_xw1(const float* __restrict__ X,
                                              const float* __restrict__ w1,
                                              float* __restrict__ outT) {
  const int lane = threadIdx.x & 31;
  const int wave = threadIdx.x >> 5;
  const int job  = blockIdx.x * 4 + wave;      // 0 .. NB*256-1
  const int b    = job >> 8;
  const int j0   = (job & 255) << 4;
  const int m    = lane & 15;
  const int h    = lane >> 4;

  v8f acc[4] = {};
  const float* Xrow = X + (size_t)(b * NNODE + j0 + m) * CH;

#pragma unroll 4
  for (int k0 = 0; k0 < CH; k0 += 4) {
    const int kb = k0 + 2 * h;
    v2f a = *(const v2f*)(Xrow + kb);          // A: lane=M row, regs/halves=K
#pragma unroll
    for (int t = 0; t < 4; ++t) {
      v2f bb;
      bb.x = w1[kb * CH + t * 16 + m];         // B: lane=N col, regs/halves=K
      bb.y = w1[(kb + 1) * CH + t * 16 + m];
      acc[t] = wmma4(a, bb, acc[t]);
    }
  }
  // D layout: acc[t][rr] = out[j0 + rr + 8h][t*16 + m]  -> store transposed
#pragma unroll
  for (int t = 0; t < 4; ++t)
#pragma unroll
    for (int rr = 0; rr < 8; ++rr)
      outT[(size_t)(b * CH + t * 16 + m) * NNODE + j0 + rr + 8 * h] =
          acc[t][rr];
}

// ---------------------------------------------------------------------------
// Kernel 1b: l = out@alpha[:64], r = out@alpha[64:], reading outT coalesced.
// ---------------------------------------------------------------------------
__global__ void k1b_lr(const float* __restrict__ outT,
                       const float* __restrict__ alpha, float* __restrict__ l,
                       float* __restrict__ r) {
  const int gid = blockIdx.x * blockDim.x + threadIdx.x;  // NB*NNODE
  const int b = gid >> 12;
  const int j = gid & (NNODE - 1);
  float aL = 0.f, aR = 0.f;
#pragma unroll 8
  for (int c = 0; c < CH; ++c) {
    float v = outT[(size_t)(b * CH + c) * NNODE + j];
    aL = fmaf(v, alpha[c], aL);
    aR = fmaf(v, alpha[CH + c], aR);
  }
  l[gid] = aL;
  r[gid] = aR;
}

// ---------------------------------------------------------------------------
// Kernel 2: for each (b, 16-row strip i0):
//   stream j; w[i,j] = adj(i,j)*exp(leaky_relu(l_i * r_j)) built in registers
//   as the WMMA A-fragment; acc(16x64) += w @ out ; track row sums;
//   normalize; fused (agg @ W2) via a second WMMA pass through LDS.
// ---------------------------------------------------------------------------
__global__ __launch_bounds__(128) void k2_main(
    const float* __restrict__ A, const float* __restrict__ outT,
    const float* __restrict__ l, const float* __restrict__ r,
    const float* __restrict__ w2, float* __restrict__ out) {
  __shared__ __align__(16) float tile[4][16 * 66];  // padded: no bank conflicts

  const int lane = threadIdx.x & 31;
  const int wave = threadIdx.x >> 5;
  const int job  = blockIdx.x * 4 + wave;      // 0 .. 1023
  const int b    = job >> 8;
  const int i0   = (job & 255) << 4;
  const int m    = lane & 15;
  const int h    = lane >> 4;

  const float l_lane = l[b * NNODE + i0 + m];
  const float* Arow  = A + (size_t)(i0 + m) * NNODE;
  const float* rb    = r + b * NNODE;
  const float* oT    = outT + (size_t)b * CH * NNODE;

  v8f acc[4] = {};
  float rsum = 0.f;

  for (int j0 = 0; j0 < NNODE; j0 += 16) {
#pragma unroll
    for (int k = 0; k < 4; ++k) {
      const int jb = j0 + 4 * k + 2 * h;
      v2f ap = *(const v2f*)(Arow + jb);
      v2f rp = *(const v2f*)(rb + jb);
      // diagonal of adj forced to 1.0
      float a0 = (i0 + m == jb)     ? 1.0f : ap.x;
      float a1 = (i0 + m == jb + 1) ? 1.0f : ap.y;
      float s0 = l_lane * rp.x;
      float s1 = l_lane * rp.y;
      s0 = s0 > 0.f ? s0 : 0.01f * s0;  // leaky_relu, jax default slope
      s1 = s1 > 0.f ? s1 : 0.01f * s1;
      v2f af;
      af.x = a0 * __expf(s0);
      af.y = a1 * __expf(s1);
      rsum += af.x + af.y;
#pragma unroll
      for (int t = 0; t < 4; ++t) {
        v2f bf = *(const v2f*)(oT + (size_t)(t * 16 + m) * NNODE + jb);
        acc[t] = wmma4(af, bf, acc[t]);
      }
    }
  }

  // complete per-row sums: lane m and lane m+16 cover disjoint j's of row m
  rsum += __shfl_xor(rsum, 16, 32);
  const float inv = 1.0f / rsum;  // valid for row (lane&15) in every lane

  // normalized agg -> LDS (D-layout to A-layout transpose within the wave)
  float* T = tile[wave];
#pragma unroll
  for (int rr = 0; rr < 8; ++rr) {
    const float invr = __shfl(inv, rr + 8 * h, 32);
#pragma unroll
    for (int t = 0; t < 4; ++t)
      T[(rr + 8 * h) * 66 + t * 16 + m] = acc[t][rr] * invr;
  }
  __syncthreads();

  // fin = agg(16x64) @ W2(64x64) via WMMA, A-fragments from LDS
  v8f fin[4] = {};
#pragma unroll 4
  for (int k0 = 0; k0 < CH; k0 += 4) {
    const int kb = k0 + 2 * h;
    v2f a = *(const v2f*)(T + m * 66 + kb);
#pragma unroll
    for (int t = 0; t < 4; ++t) {
      v2f bb;
      bb.x = w2[kb * CH + t * 16 + m];
      bb.y = w2[(kb + 1) * CH + t * 16 + m];
      fin[t] = wmma4(a, bb, fin[t]);
    }
  }

#pragma unroll
  for (int t = 0; t < 4; ++t)
#pragma unroll
    for (int rr = 0; rr < 8; ++rr)
      out[(size_t)(b * NNODE + i0 + rr + 8 * h) * CH + t * 16 + m] =
          fin[t][rr];
}

// ---------------------------------------------------------------------------
extern "C" void kernel_launch(void* const* d_in, const int* in_sizes, int n_in,
                              void* d_out, int out_size, void* d_ws,
                              size_t ws_size, hipStream_t stream) {
  const float* X     = (const float*)d_in[0];  // (B,N,64)
  const float* A     = (const float*)d_in[1];  // (N,N)
  const float* w1    = (const float*)d_in[2];  // (64,64)
  const float* w2    = (const float*)d_in[3];  // (64,64)
  const float* alpha = (const float*)d_in[4];  // (128,1)
  float* out = (float*)d_out;                  // (B,N,64) f32

  float* ws   = (float*)d_ws;
  float* outT = ws;                            // NB*CH*NNODE
  float* l    = outT + (size_t)NB * CH * NNODE;
  float* r    = l + (size_t)NB * NNODE;        // total ~4.3 MB

  k1_xw1<<<NB * 256 / 4, 128, 0, stream>>>(X, w1, outT);
  k1b_lr<<<NB * NNODE / 256, 256, 0, stream>>>(outT, alpha, l, r);
  k2_main<<<NB * 256 / 4, 128, 0, stream>>>(A, outT, l, r, w2, out);
}